// GNNBranch_40656160424194
// MI455X (gfx1250) — compile-verified
//
#include <hip/hip_runtime.h>

typedef __attribute__((ext_vector_type(2))) float v2f;
typedef __attribute__((ext_vector_type(8))) float v8f;

#define NROW 8192      // B*N
#define MSGP 68        // msg features padded 65 -> 68 (K multiple of 4)
#define MS   68        // LDS row stride, 64/68-wide activations
#define S16  20        // LDS row stride, 16-wide activations
#define S32  36        // LDS row stride, 32-wide activations

// ---------------------------------------------------------------------------
// Kernel 1: encoder MLP (16->16->32->32->64) on WMMA.
// One workgroup per 64 node-rows, 8 waves. All four layers as tiled
// 16x16x4-f32 GEMMs with LDS ping between layers; weights read from global
// (tiny, stays in WGP$/L2).  Writes msg[r][0..63]=enc, [64]=time, [65..67]=0.
// ---------------------------------------------------------------------------
__global__ void __launch_bounds__(256)
enc_kernel(const float* __restrict__ x, const float* __restrict__ timev,
           const float* __restrict__ W0, const float* __restrict__ b0,
           const float* __restrict__ W1, const float* __restrict__ b1,
           const float* __restrict__ W2, const float* __restrict__ b2,
           const float* __restrict__ W3, const float* __restrict__ b3,
           float* __restrict__ msg) {
    __shared__ float xS [64 * S16];
    __shared__ float h1S[64 * S16];
    __shared__ float h2S[64 * S32];
    __shared__ float h3S[64 * S32];

    const int tid = threadIdx.x;
    const int r0  = blockIdx.x * 64;

    for (int t = tid; t < 64 * 16; t += 256) {
        int row = t >> 4, f = t & 15;
        xS[row * S16 + f] = x[(size_t)(r0 + row) * 16 + f];
    }
    __syncthreads();

    const int wave  = tid >> 5;
    const int lane  = tid & 31;
    const int l16   = lane & 15;
    const int khalf = (lane >> 4) << 1;     // K offset 0 / 2
    const int rhalf = (lane >> 4) << 3;     // D-tile row offset 0 / 8

    // ---- layer 1: [64,16] @ W0(16x16), K=16; tiles tm=0..3 on waves 0..3 ----
    if (wave < 4) {
        const int tm = wave;
        v8f acc = {0.f, 0.f, 0.f, 0.f, 0.f, 0.f, 0.f, 0.f};
#pragma unroll
        for (int ks = 0; ks < 4; ++ks) {
            int k = ks * 4 + khalf;
            v2f a, bb;
            a.x  = xS[(tm * 16 + l16) * S16 + k];
            a.y  = xS[(tm * 16 + l16) * S16 + k + 1];
            bb.x = W0[k * 16 + l16];
            bb.y = W0[(k + 1) * 16 + l16];
            acc = __builtin_amdgcn_wmma_f32_16x16x4_f32(false, a, false, bb,
                                                        (short)0, acc, false, false);
        }
        float bias = b0[l16];
#pragma unroll
        for (int v = 0; v < 8; ++v)
            h1S[(tm * 16 + rhalf + v) * S16 + l16] = fmaxf(acc[v] + bias, 0.f);
    }
    __syncthreads();

    // ---- layer 2: [64,16] @ W1(16x32), K=16; 8 tiles, one per wave ----
    {
        const int tm = wave >> 1, tn = wave & 1, bN = tn * 16 + l16;
        v8f acc = {0.f, 0.f, 0.f, 0.f, 0.f, 0.f, 0.f, 0.f};
#pragma unroll
        for (int ks = 0; ks < 4; ++ks) {
            int k = ks * 4 + khalf;
            v2f a, bb;
            a.x  = h1S[(tm * 16 + l16) * S16 + k];
            a.y  = h1S[(tm * 16 + l16) * S16 + k + 1];
            bb.x = W1[k * 32 + bN];
            bb.y = W1[(k + 1) * 32 + bN];
            acc = __builtin_amdgcn_wmma_f32_16x16x4_f32(false, a, false, bb,
                                                        (short)0, acc, false, false);
        }
        float bias = b1[bN];
#pragma unroll
        for (int v = 0; v < 8; ++v)
            h2S[(tm * 16 + rhalf + v) * S32 + bN] = fmaxf(acc[v] + bias, 0.f);
    }
    __syncthreads();

    // ---- layer 3: [64,32] @ W2(32x32), K=32; 8 tiles, one per wave ----
    {
        const int tm = wave >> 1, tn = wave & 1, bN = tn * 16 + l16;
        v8f acc = {0.f, 0.f, 0.f, 0.f, 0.f, 0.f, 0.f, 0.f};
#pragma unroll
        for (int ks = 0; ks < 8; ++ks) {
            int k = ks * 4 + khalf;
            v2f a, bb;
            a.x  = h2S[(tm * 16 + l16) * S32 + k];
            a.y  = h2S[(tm * 16 + l16) * S32 + k + 1];
            bb.x = W2[k * 32 + bN];
            bb.y = W2[(k + 1) * 32 + bN];
            acc = __builtin_amdgcn_wmma_f32_16x16x4_f32(false, a, false, bb,
                                                        (short)0, acc, false, false);
        }
        float bias = b2[bN];
#pragma unroll
        for (int v = 0; v < 8; ++v)
            h3S[(tm * 16 + rhalf + v) * S32 + bN] = fmaxf(acc[v] + bias, 0.f);
    }
    __syncthreads();

    // ---- layer 4: [64,32] @ W3(32x64), K=32; 16 tiles, two per wave ----
    {
        const int tn = wave & 3, bN = tn * 16 + l16;
        const float bias = b3[bN];
#pragma unroll
        for (int tt = 0; tt < 2; ++tt) {
            const int tm = (wave >> 2) + 2 * tt;
            v8f acc = {0.f, 0.f, 0.f, 0.f, 0.f, 0.f, 0.f, 0.f};
#pragma unroll
            for (int ks = 0; ks < 8; ++ks) {
                int k = ks * 4 + khalf;
                v2f a, bb;
                a.x  = h3S[(tm * 16 + l16) * S32 + k];
                a.y  = h3S[(tm * 16 + l16) * S32 + k + 1];
                bb.x = W3[k * 64 + bN];
                bb.y = W3[(k + 1) * 64 + bN];
                acc = __builtin_amdgcn_wmma_f32_16x16x4_f32(false, a, false, bb,
                                                            (short)0, acc, false, false);
            }
#pragma unroll
            for (int v = 0; v < 8; ++v)     // last enc layer: no ReLU
                msg[(size_t)(r0 + tm * 16 + rhalf + v) * MSGP + bN] = acc[v] + bias;
        }
    }
    // pad columns: time + zeros
    if (tid < 64) {
        size_t row = (size_t)(r0 + tid);
        msg[row * MSGP + 64] = timev[r0 + tid];
        msg[row * MSGP + 65] = 0.f;
        msg[row * MSGP + 66] = 0.f;
        msg[row * MSGP + 67] = 0.f;
    }
}

// ---------------------------------------------------------------------------
// Kernel 2: fused edge MLP + masked aggregation.  One workgroup per (b,i),
// 16 waves; wave w owns output tile (tm=w/4, tn=w%4) of the 64x64 GEMMs.
// diff is precomputed once into LDS (shared by the 4 tn-waves per row block).
// ---------------------------------------------------------------------------
__global__ void __launch_bounds__(512)
edge_kernel(const float* __restrict__ p, const float* __restrict__ cr,
            const float* __restrict__ W1g, const float* __restrict__ b1g,
            const float* __restrict__ W2g, const float* __restrict__ b2g,
            const float* __restrict__ W3g, const float* __restrict__ b3g,
            const float* __restrict__ msg, float* __restrict__ agg) {
    __shared__ float msgS[64 * MS];      // msg rows; dead after diff -> H2 buffer
    __shared__ float diffS[64 * MS];     // msg_j - msg_i
    __shared__ float hS[64 * MS];        // H1 buffer
    __shared__ float maskS[64];
    __shared__ float aggPart[4][64];

    const int tid = threadIdx.x;
    const int r   = blockIdx.x;          // = b*64 + i
    const int b   = r >> 6;
    const int iL  = r & 63;

    const float* mbase = msg + (size_t)b * 64 * MSGP;
    for (int t = tid; t < 64 * MSGP; t += 512) msgS[t] = mbase[t];

    if (tid < 64) {
        float rad = cr[0];
        float r2  = rad * rad;
        float pix = p[(b * 64 + iL) * 2 + 0];
        float piy = p[(b * 64 + iL) * 2 + 1];
        float dx  = p[(b * 64 + tid) * 2 + 0] - pix;
        float dy  = p[(b * 64 + tid) * 2 + 1] - piy;
        maskS[tid] = (dx * dx + dy * dy <= r2) ? 1.f : 0.f;
    }
    __syncthreads();

    // diff[j][f] = msg[j][f] - msg[i][f]  (computed once, used by 4 waves)
    for (int t = tid; t < 64 * MSGP; t += 512) {
        int f = t % MSGP;
        diffS[t] = msgS[t] - msgS[iL * MS + f];
    }
    __syncthreads();

    const int wave  = tid >> 5;
    const int lane  = tid & 31;
    const int l16   = lane & 15;
    const int khalf = (lane >> 4) << 1;
    const int tm    = wave >> 2, tn = wave & 3;
    const int aRow  = tm * 16 + l16;
    const int bN    = tn * 16 + l16;
    const int row0  = tm * 16 + ((lane >> 4) << 3);

    v8f acc = {0.f, 0.f, 0.f, 0.f, 0.f, 0.f, 0.f, 0.f};

    // ---- layer 1: A = diff, B = W1 (65 real rows, padded to 68) ----
#pragma unroll
    for (int ks = 0; ks < 17; ++ks) {
        int k = ks * 4 + khalf;
        v2f a, bb;
        a.x  = diffS[aRow * MS + k];
        a.y  = diffS[aRow * MS + k + 1];
        bb.x = (k     < 65) ? W1g[k * 64 + bN]       : 0.f;
        bb.y = (k + 1 < 65) ? W1g[(k + 1) * 64 + bN] : 0.f;
        acc = __builtin_amdgcn_wmma_f32_16x16x4_f32(false, a, false, bb,
                                                    (short)0, acc, false, false);
    }
    {
        float bias = b1g[bN];
#pragma unroll
        for (int v = 0; v < 8; ++v)
            hS[(row0 + v) * MS + bN] = fmaxf(acc[v] + bias, 0.f);
    }
    __syncthreads();

    // ---- layer 2: A = H1 (hS), B = W2; output -> msgS (reuse) ----
    acc = (v8f){0.f, 0.f, 0.f, 0.f, 0.f, 0.f, 0.f, 0.f};
#pragma unroll
    for (int ks = 0; ks < 16; ++ks) {
        int k = ks * 4 + khalf;
        v2f a, bb;
        a.x  = hS[aRow * MS + k];
        a.y  = hS[aRow * MS + k + 1];
        bb.x = W2g[k * 64 + bN];
        bb.y = W2g[(k + 1) * 64 + bN];
        acc = __builtin_amdgcn_wmma_f32_16x16x4_f32(false, a, false, bb,
                                                    (short)0, acc, false, false);
    }
    {
        float bias = b2g[bN];
#pragma unroll
        for (int v = 0; v < 8; ++v)
            msgS[(row0 + v) * MS + bN] = fmaxf(acc[v] + bias, 0.f);
    }
    __syncthreads();

    // ---- layer 3: A = H2 (msgS), B = W3; no ReLU; masked reduce over j ----
    acc = (v8f){0.f, 0.f, 0.f, 0.f, 0.f, 0.f, 0.f, 0.f};
#pragma unroll
    for (int ks = 0; ks < 16; ++ks) {
        int k = ks * 4 + khalf;
        v2f a, bb;
        a.x  = msgS[aRow * MS + k];
        a.y  = msgS[aRow * MS + k + 1];
        bb.x = W3g[k * 64 + bN];
        bb.y = W3g[(k + 1) * 64 + bN];
        acc = __builtin_amdgcn_wmma_f32_16x16x4_f32(false, a, false, bb,
                                                    (short)0, acc, false, false);
    }
    {
        float bias = b3g[bN];
        float s = 0.f;
#pragma unroll
        for (int v = 0; v < 8; ++v)
            s += maskS[row0 + v] * (acc[v] + bias);
        s += __shfl_xor(s, 16, 32);              // combine M halves (same column)
        if (lane < 16) aggPart[tm][bN] = s;
    }
    __syncthreads();
    if (tid < 64)
        agg[(size_t)r * 64 + tid] =
            aggPart[0][tid] + aggPart[1][tid] + aggPart[2][tid] + aggPart[3][tid];
}

// ---------------------------------------------------------------------------
// Kernel 3: post MLP (64->64->64->64->64) as batched WMMA GEMMs.
// One workgroup per 16 rows; 4 waves, wave w owns 16x16 tile (tn = w).
// ---------------------------------------------------------------------------
__global__ void __launch_bounds__(128)
post_kernel(const float* __restrict__ agg,
            const float* __restrict__ W0, const float* __restrict__ b0,
            const float* __restrict__ W1, const float* __restrict__ b1,
            const float* __restrict__ W2, const float* __restrict__ b2,
            const float* __restrict__ W3, const float* __restrict__ b3,
            float* __restrict__ out) {
    __shared__ float hS[16 * MS];
    const int tid = threadIdx.x;
    const int r0  = blockIdx.x * 16;

    for (int t = tid; t < 16 * 64; t += 128) {
        int row = t >> 6, f = t & 63;
        hS[row * MS + f] = agg[(size_t)(r0 + row) * 64 + f];
    }
    __syncthreads();

    const int wave  = tid >> 5;                  // = tn
    const int lane  = tid & 31;
    const int l16   = lane & 15;
    const int khalf = (lane >> 4) << 1;
    const int bN    = wave * 16 + l16;
    const int row0  = (lane >> 4) << 3;

    const float* Wl[4] = {W0, W1, W2, W3};
    const float* bl[4] = {b0, b1, b2, b3};

#pragma unroll
    for (int layer = 0; layer < 4; ++layer) {
        const float* W = Wl[layer];
        v8f acc = {0.f, 0.f, 0.f, 0.f, 0.f, 0.f, 0.f, 0.f};
#pragma unroll
        for (int ks = 0; ks < 16; ++ks) {
            int k = ks * 4 + khalf;
            v2f a, bb;
            a.x  = hS[l16 * MS + k];
            a.y  = hS[l16 * MS + k + 1];
            bb.x = W[k * 64 + bN];
            bb.y = W[(k + 1) * 64 + bN];
            acc = __builtin_amdgcn_wmma_f32_16x16x4_f32(false, a, false, bb,
                                                        (short)0, acc, false, false);
        }
        float bias = bl[layer][bN];
        __syncthreads();                          // all hS reads done
        if (layer < 3) {
#pragma unroll
            for (int v = 0; v < 8; ++v)
                hS[(row0 + v) * MS + bN] = fmaxf(acc[v] + bias, 0.f);
            __syncthreads();
        } else {
#pragma unroll
            for (int v = 0; v < 8; ++v)
                out[(size_t)(r0 + row0 + v) * 64 + bN] = acc[v] + bias;
        }
    }
}

// ---------------------------------------------------------------------------
extern "C" void kernel_launch(void* const* d_in, const int* in_sizes, int n_in,
                              void* d_out, int out_size, void* d_ws, size_t ws_size,
                              hipStream_t stream) {
    const float* p   = (const float*)d_in[0];
    const float* x   = (const float*)d_in[1];
    const float* tmv = (const float*)d_in[2];
    const float* cr  = (const float*)d_in[3];
    // encoder params
    const float* eW0 = (const float*)d_in[4];  const float* eb0 = (const float*)d_in[5];
    const float* eW1 = (const float*)d_in[6];  const float* eb1 = (const float*)d_in[7];
    const float* eW2 = (const float*)d_in[8];  const float* eb2 = (const float*)d_in[9];
    const float* eW3 = (const float*)d_in[10]; const float* eb3 = (const float*)d_in[11];
    // gnn (edge) params
    const float* gW0 = (const float*)d_in[12]; const float* gb0 = (const float*)d_in[13];
    const float* gW1 = (const float*)d_in[14]; const float* gb1 = (const float*)d_in[15];
    const float* gW2 = (const float*)d_in[16]; const float* gb2 = (const float*)d_in[17];
    // post params
    const float* pW0 = (const float*)d_in[18]; const float* pb0 = (const float*)d_in[19];
    const float* pW1 = (const float*)d_in[20]; const float* pb1 = (const float*)d_in[21];
    const float* pW2 = (const float*)d_in[22]; const float* pb2 = (const float*)d_in[23];
    const float* pW3 = (const float*)d_in[24]; const float* pb3 = (const float*)d_in[25];

    float* msg = (float*)d_ws;                 // NROW * MSGP floats
    float* agg = msg + (size_t)NROW * MSGP;    // NROW * 64 floats
    float* out = (float*)d_out;

    enc_kernel<<<NROW / 64, 256, 0, stream>>>(x, tmv, eW0, eb0, eW1, eb1,
                                              eW2, eb2, eW3, eb3, msg);
    edge_kernel<<<NROW, 512, 0, stream>>>(p, cr, gW0, gb0, gW1, gb1, gW2, gb2,
                                          msg, agg);
    post_kernel<<<NROW / 16, 128, 0, stream>>>(agg, pW0, pb0, pW1, pb1,
                                               pW2, pb2, pW3, pb3, out);
}